// GraphConvolution_28587302322986
// MI455X (gfx1250) — compile-verified
//
#include <hip/hip_runtime.h>

// GCN layer: out = segment_sum(vals * (X@W)[cols], rows) + bias
// X:[100000,256] f32, W:[256,128] f32, bias:[128] f32,
// rows/cols:[3.2M] i32, vals:[3.2M] f32  ->  out:[100000,128] f32

#define N_NODES 100000
#define N_EDGES 3200000
#define IN_DIM  256
#define OUT_DIM 128
#define LDS_STRIDE (IN_DIM + 4)   // 260 floats: bank-conflict-free A-fragment reads

typedef __attribute__((ext_vector_type(2))) float v2f;
typedef __attribute__((ext_vector_type(8))) float v8f;
typedef __attribute__((ext_vector_type(4))) int   v4i;

// gfx1250 async global->LDS copy (ASYNCcnt-tracked), if the toolchain has it.
// Builtin signature (from clang diagnostic): (v4i AS1*, v4i AS3*, Imm int, Imm int)
#if __has_builtin(__builtin_amdgcn_global_load_async_to_lds_b128) && \
    __has_builtin(__builtin_amdgcn_s_wait_asynccnt)
#define USE_ASYNC_LDS 1
typedef v4i __attribute__((address_space(1)))* GPtrV4i;   // global int4*
typedef v4i __attribute__((address_space(3)))* LPtrV4i;   // LDS int4*
#endif

// ---------------------------------------------------------------------------
// Kernel 1: out[n, j] = bias[j]   (base value for the segment sum)
// ---------------------------------------------------------------------------
__global__ void __launch_bounds__(256)
gcn_bias_init_kernel(const float* __restrict__ bias, float* __restrict__ out, int total) {
    int i = blockIdx.x * 256 + threadIdx.x;
    if (i < total) out[i] = bias[i & (OUT_DIM - 1)];
}

// ---------------------------------------------------------------------------
// Kernel 2: support = X @ W via V_WMMA_F32_16X16X4_F32.
// Block = 256 threads = 8 waves; block tile = 16 rows x 128 cols.
// The 16x256 X tile (16 KB) is staged into LDS ONCE per block with
// GLOBAL_LOAD_ASYNC_TO_LDS_B128 (8x global-traffic reduction vs per-wave
// loads), then every wave feeds its A fragments from LDS (ds_load_b64).
// One wave32 per 16x16 output tile; 64 wmma steps over K=256.
// grid.x = 100000/16 = 6250 (exact, no tail -> EXEC all-ones for WMMA).
//
// ISA fragment layouts (CDNA5 ISA):
//   A 16x4 f32:  lanes 0-15 -> M=lane, VGPR0={K0 | K2}, VGPR1={K1 | K3}
//   B 4x16 f32:  row K striped across lanes (mirrored halves)
//   C/D 16x16:   VGPR r: lanes 0-15 -> M=r,N=lane ; lanes 16-31 -> M=r+8
// ---------------------------------------------------------------------------
__global__ void __launch_bounds__(256)
gcn_gemm_wmma_kernel(const float* __restrict__ X,
                     const float* __restrict__ W,
                     float* __restrict__ S) {
    __shared__ float tileX[16][LDS_STRIDE];

    const int tid  = threadIdx.x;
    const int wave = tid >> 5;           // 0..7 : N-tile index
    const int lane = tid & 31;
    const int l15  = lane & 15;
    const int hi   = lane >> 4;          // 0/1 : K-half of the fragment

    // ---- Stage X tile: 16 rows x 256 floats = 1024 float4; 4 per thread ----
    const float* __restrict__ xbase = X + (long)blockIdx.x * 16 * IN_DIM;
#pragma unroll
    for (int i = 0; i < 4; ++i) {
        const int f4 = tid + i * 256;        // float4 index 0..1023
        const int r  = f4 >> 6;              // 64 float4 per row
        const int c  = (f4 & 63) << 2;       // float column
        const float* src = xbase + r * IN_DIM + c;
        float*       dst = &tileX[r][c];
#ifdef USE_ASYNC_LDS
        __builtin_amdgcn_global_load_async_to_lds_b128(
            (GPtrV4i)src, (LPtrV4i)dst, /*offset=*/0, /*cpol=*/0);
#else
        *(float4*)dst = *(const float4*)src;
#endif
    }
#ifdef USE_ASYNC_LDS
    __builtin_amdgcn_s_wait_asynccnt(0);
#endif
    __syncthreads();

    // ---- WMMA K-loop ----
    const int col = wave * 16 + l15;                 // N position (B/D cols)
    const float* __restrict__ arow = &tileX[l15][0]; // LDS row for A fragment

    v8f acc = {};
#pragma unroll 8
    for (int k = 0; k < IN_DIM; k += 4) {
        const int kk = k + 2 * hi;
        v2f a;                                  // contiguous pair -> ds_load_b64
        a.x = arow[kk];
        a.y = arow[kk + 1];
        v2f b;                                  // two rows of W, same column
        b.x = W[kk * OUT_DIM + col];
        b.y = W[(kk + 1) * OUT_DIM + col];
        // (neg_a, A, neg_b, B, c_mod, C, reuse_a, reuse_b)
        acc = __builtin_amdgcn_wmma_f32_16x16x4_f32(
            false, a, false, b, (short)0, acc, false, false);
    }

    // Store D: VGPR r -> row (blockIdx.x*16 + r + 8*hi), col (wave*16 + l15)
    float* __restrict__ srow =
        S + ((long)blockIdx.x * 16 + 8 * hi) * OUT_DIM + wave * 16 + l15;
#pragma unroll
    for (int r = 0; r < 8; ++r) {
        srow[(long)r * OUT_DIM] = acc[r];
    }
}

// ---------------------------------------------------------------------------
// Kernel 3: scatter-add.  One wave32 per edge; lane handles 4 contiguous
// output features (float4 gather = 512B/wave coalesced; support fits in the
// 192MB L2, so gathers run at L2 bandwidth).  Native fp32 atomics
// (global_atomic_add_f32) via __hip_atomic_fetch_add — L2-side, no CAS loop.
// ---------------------------------------------------------------------------
__global__ void __launch_bounds__(256)
gcn_spmm_kernel(const int*   __restrict__ rows,
                const int*   __restrict__ cols,
                const float* __restrict__ vals,
                const float* __restrict__ S,
                float*       __restrict__ out) {
    const int edge = blockIdx.x * 8 + (threadIdx.x >> 5);
    if (edge >= N_EDGES) return;
    const int lane = threadIdx.x & 31;

    const int   r = rows[edge];
    const int   c = cols[edge];
    const float v = vals[edge];

    const float4 s4 = *(const float4*)(S + (long)c * OUT_DIM + lane * 4);
    float* o = out + (long)r * OUT_DIM + lane * 4;

    __hip_atomic_fetch_add(o + 0, v * s4.x, __ATOMIC_RELAXED, __HIP_MEMORY_SCOPE_AGENT);
    __hip_atomic_fetch_add(o + 1, v * s4.y, __ATOMIC_RELAXED, __HIP_MEMORY_SCOPE_AGENT);
    __hip_atomic_fetch_add(o + 2, v * s4.z, __ATOMIC_RELAXED, __HIP_MEMORY_SCOPE_AGENT);
    __hip_atomic_fetch_add(o + 3, v * s4.w, __ATOMIC_RELAXED, __HIP_MEMORY_SCOPE_AGENT);
}

// ---------------------------------------------------------------------------
extern "C" void kernel_launch(void* const* d_in, const int* in_sizes, int n_in,
                              void* d_out, int out_size, void* d_ws, size_t ws_size,
                              hipStream_t stream) {
    const float* X        = (const float*)d_in[0];
    const float* W        = (const float*)d_in[1];
    const float* bias     = (const float*)d_in[2];
    const int*   adj_rows = (const int*)  d_in[3];
    const int*   adj_cols = (const int*)  d_in[4];
    const float* adj_vals = (const float*)d_in[5];

    float* out = (float*)d_out;
    float* S   = (float*)d_ws;   // support: 100000*128*4B = 51.2 MB scratch

    const int total = N_NODES * OUT_DIM;
    gcn_bias_init_kernel<<<(total + 255) / 256, 256, 0, stream>>>(bias, out, total);

    gcn_gemm_wmma_kernel<<<N_NODES / 16, 256, 0, stream>>>(X, W, S);

    gcn_spmm_kernel<<<(N_EDGES + 7) / 8, 256, 0, stream>>>(
        adj_rows, adj_cols, adj_vals, S, out);
}